// cRPEncoding_14955076124952
// MI455X (gfx1250) — compile-verified
//
#include <hip/hip_runtime.h>
#include <math.h>

// ---------------------------------------------------------------------------
// Instant-NGP multires hash encoding for relative edge vectors (MI455X/gfx1250)
//
//  * Workgroup = one head h, all 3 parts, 512 points (1/thread).
//  * Per level, the three 32 KB table slices (part0..2, head h, level l)
//    stream into LDS with global_load_async_to_lds_b128, double buffered
//    (2 x 3 x 32KB = 192 KB LDS), synced with s_wait_asynccnt + barrier.
//  * Hash indices / trilinear weights computed once per (point, level),
//    reused across the 3 parts (8 ds_load_b64 gathers + 16 FMA per part).
//  * 3 x 32 f32 accumulators per thread; final stores are contiguous 128 B
//    rows per (thread, part) -> full-line write combining.
//  * BLOCK=512 -> 4 waves/SIMD -> 256-VGPR budget, no spills (640 spilled).
// ---------------------------------------------------------------------------

#ifndef __has_builtin
#define __has_builtin(x) 0
#endif

#define AS1 __attribute__((address_space(1)))
#define AS3 __attribute__((address_space(3)))

typedef int v4i __attribute__((vector_size(4 * sizeof(int))));

#if __has_builtin(__builtin_amdgcn_global_load_async_to_lds_b128)
// Signature (from hipcc diagnostic): (AS1 v4i* gsrc, AS3 v4i* ldsdst, Ii, Ii)
#define ASYNC_LOAD_B128(gp, sp)                                               \
  __builtin_amdgcn_global_load_async_to_lds_b128((AS1 v4i*)(gp),              \
                                                 (AS3 v4i*)(sp), 0, 0)
#else
#define ASYNC_LOAD_B128(gp, sp)                                               \
  do {                                                                        \
    unsigned _sa = (unsigned)(unsigned long long)(AS3 char*)(sp);             \
    unsigned long long _ga = (unsigned long long)(gp);                        \
    asm volatile("global_load_async_to_lds_b128 %0, %1, off" ::"v"(_sa),      \
                 "v"(_ga)                                                     \
                 : "memory");                                                 \
  } while (0)
#endif

#if __has_builtin(__builtin_amdgcn_s_wait_asynccnt)
#define WAIT_ASYNC0() __builtin_amdgcn_s_wait_asynccnt(0)
#else
#define WAIT_ASYNC0() asm volatile("s_wait_asynccnt 0" ::: "memory")
#endif

constexpr int N_NODES   = 10000;
constexpr int K_EDGES   = 16;
constexpr int NUM_HEADS = 8;
constexpr int DIM       = 32;          // LEVELS * FEATURE_DIM
constexpr int TSIZE     = 4096;
constexpr int FDIM      = 2;
constexpr int LEVELS    = 16;
constexpr int NPARTS    = 3;
constexpr int NPTS      = N_NODES * K_EDGES;       // 160000
constexpr int BLOCK     = 512;                     // 16 wave32 / WGP
constexpr int NBLK      = (NPTS + BLOCK - 1) / BLOCK;  // 313 (tail-guarded)
constexpr int SLICE     = TSIZE * FDIM;            // 8192 f32 = 32 KB slice
constexpr int SLICE4    = SLICE / 4;               // 2048 float4
constexpr int COPY4     = NPARTS * SLICE4;         // 6144 float4 / level-set

struct ResLevels { float r[LEVELS]; };

__device__ __forceinline__ void issue_level_copy(
    const float* __restrict__ tables, int h, int l,
    float (*dstbuf)[SLICE], int tid) {
  for (int i = tid; i < COPY4; i += BLOCK) {
    const int part = i >> 11;               // i / SLICE4
    const int e4   = i & (SLICE4 - 1);      // i % SLICE4
    const float* g = tables +
        (size_t)((part * NUM_HEADS + h) * LEVELS + l) * SLICE + (size_t)e4 * 4;
    float* s = &dstbuf[part][e4 * 4];
    ASYNC_LOAD_B128(g, s);
  }
}

__global__ void __launch_bounds__(BLOCK, 1) hashenc_kernel(
    const float* __restrict__ xyz, const int* __restrict__ edges,
    const float* __restrict__ tables, float* __restrict__ out,
    ResLevels res) {
  // 2 buffers x 3 parts x 32 KB = 192 KB (of 320 KB WGP LDS)
  __shared__ float s_tab[2][NPARTS][SLICE];

  const int      tid = threadIdx.x;
  const int      h   = blockIdx.y;
  const unsigned p   = blockIdx.x * BLOCK + tid;
  const bool     live = (p < (unsigned)NPTS);
  const unsigned pc  = live ? p : (unsigned)(NPTS - 1);  // clamp for loads
  const unsigned n   = pc >> 4;          // / K_EDGES
  const unsigned k   = pc & 15u;         // % K_EDGES
  const int      e   = edges[pc];

  const float rx = xyz[3 * n + 0] - xyz[3 * e + 0];
  const float ry = xyz[3 * n + 1] - xyz[3 * e + 1];
  const float rz = xyz[3 * n + 2] - xyz[3 * e + 2];

  float4 acc[NPARTS][DIM / 4];   // 96 floats of accumulators, stays in VGPRs

  // Prefetch level 0 into buffer 0.
  issue_level_copy(tables, h, 0, s_tab[0], tid);

#pragma unroll
  for (int l = 0; l < LEVELS; ++l) {
    const int cur = l & 1;
    WAIT_ASYNC0();       // this wave's slice-l async loads have landed in LDS
    __syncthreads();     // all waves' loads landed; buf^1 reads from l-1 done
    if (l + 1 < LEVELS)  // stream next level while computing this one
      issue_level_copy(tables, h, l + 1, s_tab[cur ^ 1], tid);

    // --- hash indices + trilinear weights, once per (point, level) ---
    const float r  = res.r[l];
    const float sx = rx * r, sy = ry * r, sz = rz * r;
    const float bx = floorf(sx), by = floorf(sy), bz = floorf(sz);
    const float fx = sx - bx, fy = sy - by, fz = sz - bz;
    const unsigned hx0 = (unsigned)(int)bx;                     // prime 1
    const unsigned hy0 = (unsigned)(int)by * 2654435761u;
    const unsigned hz0 = (unsigned)(int)bz * 805459861u;
    const unsigned hx1 = hx0 + 1u;
    const unsigned hy1 = hy0 + 2654435761u;
    const unsigned hz1 = hz0 + 805459861u;
    const unsigned m   = TSIZE - 1;
    const unsigned i00 = hx0 ^ hy0, i01 = hx0 ^ hy1;
    const unsigned i10 = hx1 ^ hy0, i11 = hx1 ^ hy1;
    // corner c = 4*x + 2*y + z  (meshgrid 'ij' order of the reference)
    int id[8];
    id[0] = (i00 ^ hz0) & m;  id[1] = (i00 ^ hz1) & m;
    id[2] = (i01 ^ hz0) & m;  id[3] = (i01 ^ hz1) & m;
    id[4] = (i10 ^ hz0) & m;  id[5] = (i10 ^ hz1) & m;
    id[6] = (i11 ^ hz0) & m;  id[7] = (i11 ^ hz1) & m;
    const float gx = 1.f - fx, gy = 1.f - fy, gz = 1.f - fz;
    const float w00 = gx * gy, w01 = gx * fy, w10 = fx * gy, w11 = fx * fy;
    float w[8];
    w[0] = w00 * gz; w[1] = w00 * fz; w[2] = w01 * gz; w[3] = w01 * fz;
    w[4] = w10 * gz; w[5] = w10 * fz; w[6] = w11 * gz; w[7] = w11 * fz;

    // --- gather + interpolate for all 3 parts from LDS (ds_load_b64) ---
#pragma unroll
    for (int part = 0; part < NPARTS; ++part) {
      const float2* tb2 = reinterpret_cast<const float2*>(s_tab[cur][part]);
      float a0 = 0.f, a1 = 0.f;
#pragma unroll
      for (int c = 0; c < 8; ++c) {
        const float2 f = tb2[id[c]];
        a0 = fmaf(w[c], f.x, a0);
        a1 = fmaf(w[c], f.y, a1);
      }
      float4& d4 = acc[part][l >> 1];
      if (l & 1) { d4.z = a0; d4.w = a1; } else { d4.x = a0; d4.y = a1; }
    }
  }

  // Each live thread writes 3 contiguous 128 B rows: out[part][n][h][k][:].
  if (live) {
#pragma unroll
    for (int part = 0; part < NPARTS; ++part) {
      const size_t base =
          ((size_t)((part * N_NODES + n) * NUM_HEADS + h) * K_EDGES + k) * DIM;
      float4* o4 = reinterpret_cast<float4*>(out + base);
#pragma unroll
      for (int j = 0; j < DIM / 4; ++j) o4[j] = acc[part][j];
    }
  }
}

extern "C" void kernel_launch(void* const* d_in, const int* in_sizes, int n_in,
                              void* d_out, int out_size, void* d_ws,
                              size_t ws_size, hipStream_t stream) {
  const float* xyz    = (const float*)d_in[0];
  const int*   edges  = (const int*)d_in[1];   // JAX canonicalizes i64 -> i32
  const float* tables = (const float*)d_in[2];
  float*       out    = (float*)d_out;

  // Geometric resolution ladder computed in double (matches numpy ** in f64).
  ResLevels res;
  const double B = pow(1.0e7, 1.0 / (double)(LEVELS - 1));
  for (int l = 0; l < LEVELS; ++l) res.r[l] = (float)pow(B, (double)l);

  dim3 grid(NBLK, NUM_HEADS);  // (313, 8)
  hashenc_kernel<<<grid, BLOCK, 0, stream>>>(xyz, edges, tables, out, res);
}